// GCN_58660663329125
// MI455X (gfx1250) — compile-verified
//
#include <hip/hip_runtime.h>
#include <hip/hip_bf16.h>
#include <math.h>

typedef float v2f __attribute__((ext_vector_type(2)));
typedef float v8f __attribute__((ext_vector_type(8)));

// ---------------- degree / normalization ----------------

__global__ void k_init_deg(unsigned* __restrict__ deg, int n) {
  int i = blockIdx.x * blockDim.x + threadIdx.x;
  if (i < n) deg[i] = 1u;  // self-loop
}

__global__ void k_deg(unsigned* __restrict__ deg, const long long* __restrict__ ei, int E) {
  int i = blockIdx.x * blockDim.x + threadIdx.x;
  if (i < E) atomicAdd(&deg[(int)ei[(long long)E + i]], 1u);
}

__global__ void k_dinv(const unsigned* __restrict__ deg, float* __restrict__ dinv, int n) {
  int i = blockIdx.x * blockDim.x + threadIdx.x;
  if (i < n) dinv[i] = rsqrtf((float)deg[i]);
}

// ---------------- WMMA fp32 GEMM: C[nrows,NOUT] = A[nrows,K] @ B[K,NOUT] ----------------
// One wave computes a 16 x (NTILES*16) tile using V_WMMA_F32_16X16X4_F32.
// A-frag layout (ISA 7.12.2, 32-bit A 16x4): lane<16 -> K=0,1 ; lane>=16 -> K=2,3 ; M=lane&15.
// C/D layout: VGPR r, lane<16 -> M=r,N=lane ; lane>=16 -> M=r+8,N=lane-16.

template<int K, int NOUT, int NTILES>
__global__ void k_gemm_wmma(const float* __restrict__ A, const float* __restrict__ B,
                            float* __restrict__ C, int nrows) {
  int wave = (blockIdx.x * blockDim.x + threadIdx.x) >> 5;
  int lane = threadIdx.x & 31;
  int mbase = wave * 16;
  if (mbase >= nrows) return;           // wave-uniform: EXEC stays all-1 for WMMA
  int lm = lane & 15;
  int hi = lane >> 4;
  int kg = hi * 2;

  v8f acc[NTILES];
#pragma unroll
  for (int t = 0; t < NTILES; ++t) acc[t] = (v8f){0.f,0.f,0.f,0.f,0.f,0.f,0.f,0.f};

  const float* __restrict__ arow = A + (size_t)(mbase + lm) * K;
#pragma unroll 4
  for (int k = 0; k < K; k += 4) {
    v2f a;
    a.x = arow[k + kg];
    a.y = arow[k + kg + 1];
#pragma unroll
    for (int t = 0; t < NTILES; ++t) {
      int n = t * 16 + lm;
      v2f b;
      bool ok = (n < NOUT);
      b.x = ok ? B[(k + kg) * NOUT + n] : 0.0f;
      b.y = ok ? B[(k + kg + 1) * NOUT + n] : 0.0f;
      acc[t] = __builtin_amdgcn_wmma_f32_16x16x4_f32(
          /*neg_a=*/false, a, /*neg_b=*/false, b,
          /*c_mod=*/(short)0, acc[t], /*reuse_a=*/false, /*reuse_b=*/false);
    }
  }

#pragma unroll
  for (int t = 0; t < NTILES; ++t) {
    int n = t * 16 + lm;
    if (n < NOUT) {
#pragma unroll
      for (int r = 0; r < 8; ++r) {
        int m = mbase + r + hi * 8;
        C[(size_t)m * NOUT + n] = acc[t][r];
      }
    }
  }
}

// ---------------- accumulator init: agg = b + h * dinv^2 (self-loop term) ----------------

template<int F>
__global__ void k_bias_self(float* __restrict__ agg, const float* __restrict__ h,
                            const float* __restrict__ dinv, const float* __restrict__ b, int n) {
  int idx = blockIdx.x * blockDim.x + threadIdx.x;
  if (idx >= n * F) return;
  int i = idx / F;
  int f = idx - i * F;
  float dv = dinv[i];
  agg[idx] = b[f] + h[idx] * dv * dv;
}

// ---------------- edge scatter: one wave per edge, lanes over features ----------------

template<int F>
__global__ void k_scatter(float* __restrict__ agg, const float* __restrict__ h,
                          const float* __restrict__ dinv, const long long* __restrict__ ei,
                          int E) {
  int wave = (blockIdx.x * blockDim.x + threadIdx.x) >> 5;
  int lane = threadIdx.x & 31;
  if (wave >= E) return;
  int s = (int)ei[wave];
  int d = (int)ei[(long long)E + wave];
  float nrm = dinv[s] * dinv[d];
  const float* __restrict__ hs = h + (size_t)s * F;
  float* __restrict__ ad = agg + (size_t)d * F;
  unsafeAtomicAdd(&ad[lane], hs[lane] * nrm);
  int f2 = lane + 32;
  if (f2 < F) unsafeAtomicAdd(&ad[f2], hs[f2] * nrm);
}

// ---------------- elementwise ReLU ----------------

__global__ void k_relu(float* __restrict__ p, int n) {
  int i = blockIdx.x * blockDim.x + threadIdx.x;
  if (i < n) p[i] = fmaxf(p[i], 0.0f);
}

// ---------------- log_softmax over 40 classes, one wave per node ----------------

__global__ void k_logsoftmax(float* __restrict__ out, int n) {
  int wave = (blockIdx.x * blockDim.x + threadIdx.x) >> 5;
  int lane = threadIdx.x & 31;
  if (wave >= n) return;
  float* row = out + (size_t)wave * 40;
  float v0 = row[lane];                                  // lanes 0..31 cover f 0..31
  float v1 = (lane < 8) ? row[32 + lane] : -INFINITY;    // lanes 0..7 cover f 32..39
  float m = fmaxf(v0, v1);
#pragma unroll
  for (int off = 16; off > 0; off >>= 1) m = fmaxf(m, __shfl_xor(m, off, 32));
  float s = expf(v0 - m) + ((lane < 8) ? expf(v1 - m) : 0.0f);
#pragma unroll
  for (int off = 16; off > 0; off >>= 1) s += __shfl_xor(s, off, 32);
  float lse = m + logf(s);
  row[lane] = v0 - lse;
  if (lane < 8) row[32 + lane] = v1 - lse;
}

// ---------------- launch ----------------

extern "C" void kernel_launch(void* const* d_in, const int* in_sizes, int n_in,
                              void* d_out, int out_size, void* d_ws, size_t ws_size,
                              hipStream_t stream) {
  const float*      x  = (const float*)d_in[0];
  const long long*  ei = (const long long*)d_in[1];   // int64 edge_index [2, E]
  const float*      W1 = (const float*)d_in[2];
  const float*      b1 = (const float*)d_in[3];
  const float*      W2 = (const float*)d_in[4];
  const float*      b2 = (const float*)d_in[5];
  float*            out = (float*)d_out;

  const int Fin = 128, H = 64, Cc = 40;
  int N = in_sizes[0] / Fin;     // 100000
  int E = in_sizes[1] / 2;       // 1600000

  // workspace layout (floats): dinv[N] | deg[N] | h[N*64] | agg1[N*64]
  float*    wsf  = (float*)d_ws;
  float*    dinv = wsf;
  unsigned* deg  = (unsigned*)(wsf + N);
  float*    h1   = wsf + 2 * (size_t)N;
  float*    agg1 = h1 + (size_t)N * H;
  float*    h2   = h1;                 // reuse h1 after layer-1 scatter completes

  const int T = 256;                   // 8 waves/block (wave32)
  int nb_N   = (N + T - 1) / T;
  int nb_E   = (E + T - 1) / T;
  int nb_NH  = (N * H + T - 1) / T;
  int nb_NC  = (N * Cc + T - 1) / T;
  int mtiles = (N + 15) / 16;
  int nb_gemm = (mtiles + 7) / 8;      // 8 waves per block, 1 M-tile per wave
  int nb_edgewave = (E + 7) / 8;       // 1 edge per wave
  int nb_nodewave = (N + 7) / 8;       // 1 node per wave

  // normalization: deg (with self-loop), dinv = rsqrt(deg)
  k_init_deg<<<nb_N, T, 0, stream>>>(deg, N);
  k_deg<<<nb_E, T, 0, stream>>>(deg, ei, E);
  k_dinv<<<nb_N, T, 0, stream>>>(deg, dinv, N);

  // layer 1: h1 = x @ W1 ; agg1 = b1 + self + edge scatter ; ReLU
  k_gemm_wmma<128, 64, 4><<<nb_gemm, T, 0, stream>>>(x, W1, h1, N);
  k_bias_self<64><<<nb_NH, T, 0, stream>>>(agg1, h1, dinv, b1, N);
  k_scatter<64><<<nb_edgewave, T, 0, stream>>>(agg1, h1, dinv, ei, E);
  k_relu<<<nb_NH, T, 0, stream>>>(agg1, N * H);

  // layer 2: h2 = relu(agg1) @ W2 ; out = b2 + self + edge scatter ; log_softmax
  k_gemm_wmma<64, 40, 3><<<nb_gemm, T, 0, stream>>>(agg1, W2, h2, N);
  k_bias_self<40><<<nb_NC, T, 0, stream>>>(out, h2, dinv, b2, N);
  k_scatter<40><<<nb_edgewave, T, 0, stream>>>(out, h2, dinv, ei, E);
  k_logsoftmax<<<nb_nodewave, T, 0, stream>>>(out, N);
}